// deltaEColorLoss_24489903522093
// MI455X (gfx1250) — compile-verified
//
#include <hip/hip_runtime.h>
#include <stdint.h>

// deltaE (CIEDE2000) mean over two (B,3,H,W) f32 images, reinterpreted as
// contiguous float3 pixels (reference does a plain reshape, not a transpose).
//
// MI455X plan:
//  - Transcendental-VALU bound (~40 trans ops/pixel * 8.4M pixels); memory floor
//    is only ~8.6us (201MB @ 23.3 TB/s). Use hw v_exp/v_log/v_sin/v_cos/v_rsq.
//  - CDNA5 data path: per-wave double-buffered global_load_async_to_lds_b128
//    (ASYNCcnt + s_wait_asynccnt), wave-private LDS slices -> no block barriers.
//  - Deterministic two-pass reduction (block partials in d_ws, then finalize).

#define BLOCK_THREADS 256
#define WAVES_PER_BLOCK (BLOCK_THREADS / 32)
#define TILE_PIX 128                 // pixels per wave-tile
#define TILE_F (TILE_PIX * 3)        // 384 floats per image per wave-tile
#define GRID_BLOCKS 1024

// ---------------------------------------------------------------- color math
__device__ __forceinline__ float srgb2lin(float c) {
  // pow(x, 2.4) via hw log/exp
  return (c <= 0.04045f) ? c * (1.0f / 12.92f)
                         : __expf(2.4f * __logf((c + 0.055f) * (1.0f / 1.055f)));
}

__device__ __forceinline__ float labf(float t) {
  // cbrt(t) for t > 0.008856 via hw log/exp
  return (t > 0.008856f) ? __expf((1.0f / 3.0f) * __logf(t))
                         : 7.787f * t + (16.0f / 116.0f);
}

__device__ __forceinline__ void rgb2lab(float r, float g, float b,
                                        float& L, float& A, float& B) {
  float lr = srgb2lin(r), lg = srgb2lin(g), lb = srgb2lin(b);
  float x = (0.412453f * lr + 0.357580f * lg + 0.180423f * lb) * (1.0f / 0.95047f);
  float y = (0.212671f * lr + 0.715160f * lg + 0.072169f * lb);
  float z = (0.019334f * lr + 0.119193f * lg + 0.950227f * lb) * (1.0f / 1.08883f);
  float fx = labf(x), fy = labf(y), fz = labf(z);
  L = 116.0f * fy - 16.0f;
  A = 500.0f * (fx - fy);
  B = 200.0f * (fy - fz);
}

__device__ __forceinline__ float pow7f(float x) {
  float x2 = x * x;
  float x4 = x2 * x2;
  return x4 * x2 * x;
}

__device__ __forceinline__ float ciede2000(float L1, float a1, float b1,
                                           float L2, float a2, float b2) {
  const float P257 = 6103515625.0f;               // 25^7
  const float R2D = 57.29577951308232f;
  const float D2R = 0.017453292519943295f;

  float C1 = __fsqrt_rn(a1 * a1 + b1 * b1);
  float C2 = __fsqrt_rn(a2 * a2 + b2 * b2);
  float Cbar = 0.5f * (C1 + C2);
  float c7 = pow7f(Cbar);
  float G = 0.5f * (1.0f - __fsqrt_rn(__fdividef(c7, c7 + P257)));
  float a1p = a1 * (1.0f + G);
  float a2p = a2 * (1.0f + G);
  float C1p = __fsqrt_rn(a1p * a1p + b1 * b1);
  float C2p = __fsqrt_rn(a2p * a2p + b2 * b2);

  float h1 = atan2f(b1, a1p) * R2D; if (h1 < 0.0f) h1 += 360.0f;
  float h2 = atan2f(b2, a2p) * R2D; if (h2 < 0.0f) h2 += 360.0f;

  float dLp = L2 - L1;
  float dCp = C2p - C1p;
  float CC = C1p * C2p;

  float hd = h2 - h1;
  float dh = (hd > 180.0f) ? hd - 360.0f : ((hd < -180.0f) ? hd + 360.0f : hd);
  if (CC == 0.0f) dh = 0.0f;
  float dHp = 2.0f * __fsqrt_rn(CC) * __sinf(0.5f * dh * D2R);

  float Lbar = 0.5f * (L1 + L2);
  float Cbarp = 0.5f * (C1p + C2p);
  float hsum = h1 + h2;
  float hbar = (fabsf(h1 - h2) <= 180.0f)
                   ? 0.5f * hsum
                   : ((hsum < 360.0f) ? 0.5f * (hsum + 360.0f)
                                      : 0.5f * (hsum - 360.0f));
  if (CC == 0.0f) hbar = hsum;

  float hr = hbar * D2R;
  float T = 1.0f - 0.17f * __cosf(hr - 30.0f * D2R)
                 + 0.24f * __cosf(2.0f * hr)
                 + 0.32f * __cosf(3.0f * hr + 6.0f * D2R)
                 - 0.20f * __cosf(4.0f * hr - 63.0f * D2R);

  float Lm50 = Lbar - 50.0f;
  float Lm50sq = Lm50 * Lm50;
  float Sl = 1.0f + 0.015f * Lm50sq * __frsqrt_rn(20.0f + Lm50sq);
  float Sc = 1.0f + 0.045f * Cbarp;
  float Sh = 1.0f + 0.015f * Cbarp * T;

  float u = (hbar - 275.0f) * (1.0f / 25.0f);
  float dtheta = 30.0f * __expf(-(u * u));
  float cb7 = pow7f(Cbarp);
  float Rc = 2.0f * __fsqrt_rn(__fdividef(cb7, cb7 + P257));
  float Rt = -Rc * __sinf(2.0f * dtheta * D2R);

  float tL = __fdividef(dLp, Sl);
  float tC = __fdividef(dCp, Sc);
  float tH = __fdividef(dHp, Sh);
  float s = tL * tL + tC * tC + tH * tH + Rt * tC * tH;
  return __fsqrt_rn(fmaxf(s, 0.0f));
}

// -------------------------------------------------------- CDNA5 async copies
__device__ __forceinline__ uint32_t lds_byte_off(const void* p) {
  // Generic LDS address: high 32 bits = shared aperture, low 32 = LDS offset.
  return (uint32_t)(uintptr_t)p;
}

// Copy one wave-tile (128 pixels) of both images into this wave's LDS slice.
// 3 b128 loads per image per lane -> 6 async instructions per tile per wave.
__device__ __forceinline__ void async_tile_copy(const float* __restrict__ gen,
                                                const float* __restrict__ gt,
                                                float (&buf)[2][TILE_F],
                                                long long tile, int lane) {
  // WAR safety: make sure our prior ds_loads on this buffer have completed
  // before the async unit starts overwriting it.
  asm volatile("s_wait_dscnt 0" ::: "memory");
  long long fbase = tile * (long long)TILE_F;
#pragma unroll
  for (int k = 0; k < 3; ++k) {
    int f4 = lane + 32 * k;                     // float4 index within tile
    uint32_t l0 = lds_byte_off(&buf[0][f4 * 4]);
    uint32_t l1 = lds_byte_off(&buf[1][f4 * 4]);
    const float* g0 = gen + fbase + (long long)(f4 * 4);
    const float* g1 = gt  + fbase + (long long)(f4 * 4);
    asm volatile("global_load_async_to_lds_b128 %0, %1, off"
                 :: "v"(l0), "v"(g0) : "memory");
    asm volatile("global_load_async_to_lds_b128 %0, %1, off"
                 :: "v"(l1), "v"(g1) : "memory");
  }
}

__device__ __forceinline__ float tile_compute(const float (&buf)[2][TILE_F],
                                              int lane) {
  float acc = 0.0f;
#pragma unroll
  for (int j = 0; j < 4; ++j) {
    int q = lane + 32 * j;                      // pixel within tile
    // buf[0] = genImage pixels, buf[1] = gtImage pixels
    float r2 = buf[0][3 * q + 0], g2 = buf[0][3 * q + 1], b2 = buf[0][3 * q + 2];
    float r1 = buf[1][3 * q + 0], g1 = buf[1][3 * q + 1], b1 = buf[1][3 * q + 2];
    float L1, A1, B1, L2, A2, B2;
    rgb2lab(r1, g1, b1, L1, A1, B1);            // gt -> (L1,a1,b1)
    rgb2lab(r2, g2, b2, L2, A2, B2);            // gen -> (L2,a2,b2)
    acc += fabsf(ciede2000(L1, A1, B1, L2, A2, B2));
  }
  return acc;
}

// ------------------------------------------------------------------- kernels
__global__ __launch_bounds__(BLOCK_THREADS)
void deltaE_main_kernel(const float* __restrict__ gen,
                        const float* __restrict__ gt,
                        float* __restrict__ partial,
                        long long npix) {
  __shared__ float lds[2][WAVES_PER_BLOCK][2][TILE_F];   // 48 KB
  __shared__ float red[BLOCK_THREADS];

  const int tid = threadIdx.x;
  const int lane = tid & 31;
  const int w = tid >> 5;

  const long long ntiles = npix / TILE_PIX;
  const long long wave_gid = (long long)blockIdx.x * WAVES_PER_BLOCK + w;
  const long long wave_stride = (long long)gridDim.x * WAVES_PER_BLOCK;

  float acc = 0.0f;

  long long t = wave_gid;
  if (t < ntiles) {
    int buf = 0;
    async_tile_copy(gen, gt, lds[0][w], t, lane);        // prologue prefetch
    for (; t < ntiles; t += wave_stride) {
      long long tn = t + wave_stride;
      if (tn < ntiles) {
        async_tile_copy(gen, gt, lds[buf ^ 1][w], tn, lane);
        // wait for the CURRENT tile's 6 copies (the 6 just issued may remain)
        asm volatile("s_wait_asynccnt 6" ::: "memory");
      } else {
        asm volatile("s_wait_asynccnt 0" ::: "memory");
      }
      acc += tile_compute(lds[buf][w], lane);
      buf ^= 1;
    }
  }

  // Scalar tail for npix % TILE_PIX (zero for the reference shape).
  for (long long p = ntiles * TILE_PIX + (long long)blockIdx.x * BLOCK_THREADS + tid;
       p < npix; p += (long long)gridDim.x * BLOCK_THREADS) {
    float r2 = gen[3 * p + 0], g2 = gen[3 * p + 1], b2 = gen[3 * p + 2];
    float r1 = gt[3 * p + 0],  g1 = gt[3 * p + 1],  b1 = gt[3 * p + 2];
    float L1, A1, B1, L2, A2, B2;
    rgb2lab(r1, g1, b1, L1, A1, B1);
    rgb2lab(r2, g2, b2, L2, A2, B2);
    acc += fabsf(ciede2000(L1, A1, B1, L2, A2, B2));
  }

  // Deterministic block reduction -> one partial per block.
  red[tid] = acc;
  __syncthreads();
  for (int s = BLOCK_THREADS / 2; s > 0; s >>= 1) {
    if (tid < s) red[tid] += red[tid + s];
    __syncthreads();
  }
  if (tid == 0) partial[blockIdx.x] = red[0];
}

__global__ __launch_bounds__(256)
void deltaE_finalize_kernel(const float* __restrict__ partial, int n,
                            float* __restrict__ out, float inv_npix) {
  __shared__ float red[256];
  float s = 0.0f;
  for (int i = threadIdx.x; i < n; i += 256) s += partial[i];
  red[threadIdx.x] = s;
  __syncthreads();
  for (int k = 128; k > 0; k >>= 1) {
    if (threadIdx.x < k) red[threadIdx.x] += red[threadIdx.x + k];
    __syncthreads();
  }
  if (threadIdx.x == 0) out[0] = red[0] * inv_npix;
}

// -------------------------------------------------------------------- launch
extern "C" void kernel_launch(void* const* d_in, const int* in_sizes, int n_in,
                              void* d_out, int out_size, void* d_ws, size_t ws_size,
                              hipStream_t stream) {
  const float* gen = (const float*)d_in[0];   // genImage (B,3,H,W) f32
  const float* gt  = (const float*)d_in[1];   // gtImage  (B,3,H,W) f32
  float* out = (float*)d_out;
  float* partial = (float*)d_ws;

  const long long total_f = (long long)in_sizes[0];   // B*3*H*W
  const long long npix = total_f / 3;

  int nblocks = GRID_BLOCKS;
  size_t max_partials = ws_size / sizeof(float);
  if ((size_t)nblocks > max_partials && max_partials > 0)
    nblocks = (int)max_partials;
  if (nblocks < 1) nblocks = 1;

  deltaE_main_kernel<<<nblocks, BLOCK_THREADS, 0, stream>>>(gen, gt, partial, npix);
  deltaE_finalize_kernel<<<1, 256, 0, stream>>>(partial, nblocks, out,
                                                1.0f / (float)npix);
}